// VectorQuantizer2_7481833030165
// MI455X (gfx1250) — compile-verified
//
#include <hip/hip_runtime.h>

typedef __attribute__((ext_vector_type(2))) float v2f;
typedef __attribute__((ext_vector_type(8))) float v8f;

#define NUMEL (4*32*8*32*32)   // B*C*T*H*W = 1048576
#define CCH   32               // channels
#define NV    4096             // codebook size
#define TFR   32               // B*T frames

// ---- CDNA5 async global->LDS staging (guarded so compile never regresses) ----
#if defined(__has_builtin)
#if __has_builtin(__builtin_amdgcn_global_load_async_to_lds_b128)
#define HAVE_ASYNC 1
#endif
#endif
#ifndef HAVE_ASYNC
#define HAVE_ASYNC 0
#endif

#if HAVE_ASYNC
typedef int v4i_ls __attribute__((vector_size(16)));            // int __vector(4)
typedef __attribute__((address_space(1))) v4i_ls* gas1p;        // global src
typedef __attribute__((address_space(3))) v4i_ls* las3p;        // LDS dst
#if __has_builtin(__builtin_amdgcn_s_wait_asynccnt)
#define WAIT_ASYNC(n) __builtin_amdgcn_s_wait_asynccnt(n)
#else
#define WAIT_ASYNC(n) asm volatile("s_wait_asynccnt %0" ::"i"(n) : "memory")
#endif
#else
#define WAIT_ASYNC(n)
#endif

// ---------------- init: f_rest = f, f_hat = 0, loss = 0 ----------------
__global__ void vq_init(const float* __restrict__ f, float* __restrict__ f_rest,
                        float* __restrict__ f_hat, float* __restrict__ loss) {
  int i = blockIdx.x * blockDim.x + threadIdx.x;
  if (i < NUMEL) { f_rest[i] = f[i]; f_hat[i] = 0.f; }
  if (i == 0) loss[0] = 0.f;
}

// ---------------- e_sq[v] = sum_k emb[v][k]^2 ----------------
__global__ void vq_esq(const float* __restrict__ emb, float* __restrict__ esq) {
  int v = blockIdx.x * blockDim.x + threadIdx.x;
  if (v >= NV) return;
  const float* e = emb + v * CCH;
  float s = 0.f;
#pragma unroll
  for (int k = 0; k < CCH; ++k) s += e[k] * e[k];
  esq[v] = s;
}

// ---------- bilinear downsample f_rest [B,C,T,32,32] -> X [N,32], N=32*pn*pn ----------
// identity when pn==32 (source coords degenerate to integers, w=0)
__global__ void vq_down(const float* __restrict__ f_rest, float* __restrict__ X, int pn) {
  int gid = blockIdx.x * blockDim.x + threadIdx.x;
  int N = TFR * pn * pn;
  if (gid >= N * CCH) return;
  int c = gid & 31;
  int n = gid >> 5;
  int pw = n % pn;
  int ph = (n / pn) % pn;
  int bt = n / (pn * pn);
  int b = bt >> 3, t = bt & 7;
  float scale = 32.f / (float)pn;
  float sy = fminf(fmaxf((ph + 0.5f) * scale - 0.5f, 0.f), 31.f);
  float sx = fminf(fmaxf((pw + 0.5f) * scale - 0.5f, 0.f), 31.f);
  int y0 = (int)sy; int y1 = min(y0 + 1, 31); float wy = sy - (float)y0;
  int x0 = (int)sx; int x1 = min(x0 + 1, 31); float wx = sx - (float)x0;
  const float* base = f_rest + ((b * CCH + c) * 8 + t) * 1024;
  float v00 = base[y0 * 32 + x0], v01 = base[y0 * 32 + x1];
  float v10 = base[y1 * 32 + x0], v11 = base[y1 * 32 + x1];
  X[n * CCH + c] = (v00 * (1.f - wy) + v10 * wy) * (1.f - wx) +
                   (v01 * (1.f - wy) + v11 * wy) * wx;
}

// ---------- WMMA argmin with async-staged, double-buffered LDS codebook ----------
// d(n,v) = esq[v] - 2 * dot(X[n], emb[v])   (||x||^2 dropped: constant per row)
// block = 8 waves; each wave owns one 16-row tile; emb staged in 128-code chunks.
#define CHUNK    128
#define NCHUNK   (NV / CHUNK)       // 32
#define ROWPITCH 36                 // floats per code row in LDS (144B: 16B-aligned,
                                    // conflict-free across the 16 fragment lanes)

__global__ __launch_bounds__(256) void vq_argmin(const float* __restrict__ X,
                                                 const float* __restrict__ emb,
                                                 const float* __restrict__ esq,
                                                 int* __restrict__ idx, int Nt) {
  __shared__ __align__(16) float eb[2 * CHUNK * ROWPITCH];   // 36 KB double buffer
  int tid = threadIdx.x;
  int wave = tid >> 5, lane = tid & 31;
  int mtile = blockIdx.x * 8 + wave;
  bool haveRows = (mtile < Nt);          // wave-uniform; EXEC stays all-1s for WMMA
  int ln = lane & 15;
  int koff = (lane >> 4) * 2;            // f32 16x16x4 fragment K sub-offset

  // A fragments: 8 k-steps cover K=32 (lane holds row mtile*16+ln, k = 4s+koff+{0,1})
  v2f a[8];
  if (haveRows) {
    const float* xr = X + (mtile * 16 + ln) * CCH + koff;
#pragma unroll
    for (int s = 0; s < 8; ++s) { a[s].x = xr[4 * s]; a[s].y = xr[4 * s + 1]; }
  } else {
#pragma unroll
    for (int s = 0; s < 8; ++s) { a[s].x = 0.f; a[s].y = 0.f; }
  }

  float minv[8]; int mini[8];
#pragma unroll
  for (int j = 0; j < 8; ++j) { minv[j] = 3.4e38f; mini[j] = 0; }

  // stage one 128-code chunk (16KB) into LDS buffer `buf`: 1024 b128 ops, 4/thread
  auto stage = [&](int ch, int buf) {
    const float* src0 = emb + (size_t)ch * CHUNK * CCH;
#pragma unroll
    for (int q = 0; q < 4; ++q) {
      int item = tid + q * 256;          // 0..1023
      int row = item >> 3, seg = item & 7;
      const float* g = src0 + row * CCH + seg * 4;
      float* l = eb + buf * (CHUNK * ROWPITCH) + row * ROWPITCH + seg * 4;
#if HAVE_ASYNC
      __builtin_amdgcn_global_load_async_to_lds_b128(
          (gas1p)(uintptr_t)g, (las3p)(uintptr_t)l, 0, 0);
#else
      *(float4*)l = *(const float4*)g;
#endif
    }
  };

  stage(0, 0);
  for (int ch = 0; ch < NCHUNK; ++ch) {
    int cur = ch & 1;
    if (ch + 1 < NCHUNK) {               // prefetch next chunk while computing this one
      stage(ch + 1, cur ^ 1);
      WAIT_ASYNC(4);                     // oldest 4 async ops (= chunk ch) complete
    } else {
      WAIT_ASYNC(0);
    }
    __syncthreads();                     // chunk ch visible to all waves

    if (haveRows) {
      const float* bb = eb + cur * (CHUNK * ROWPITCH);
      for (int vt = 0; vt < CHUNK / 16; ++vt) {
        int colL = vt * 16 + ln;                       // this lane's code (local)
        const float* er = bb + colL * ROWPITCH + koff; // ds_load_b64 feed
        v8f c = {0.f, 0.f, 0.f, 0.f, 0.f, 0.f, 0.f, 0.f};
#pragma unroll
        for (int s = 0; s < 8; ++s) {
          v2f b = *(const v2f*)(er + 4 * s);
          c = __builtin_amdgcn_wmma_f32_16x16x4_f32(false, a[s], false, b,
                                                    (short)0, c, false, false);
        }
        int col = ch * CHUNK + colL;
        float es = esq[col];
#pragma unroll
        for (int j = 0; j < 8; ++j) {
          float d = es - 2.f * c[j];
          if (d < minv[j]) { minv[j] = d; mini[j] = col; }
        }
      }
    }
    __syncthreads();                     // all waves done reading before buffer reuse
  }

  if (!haveRows) return;
  // reduce over the 16 lanes sharing a row (xor stays within each half-wave)
#pragma unroll
  for (int j = 0; j < 8; ++j) {
    float mv = minv[j]; int mi = mini[j];
#pragma unroll
    for (int off = 1; off < 16; off <<= 1) {
      float ov = __shfl_xor(mv, off, 32);
      int   oi = __shfl_xor(mi, off, 32);
      if (ov < mv || (ov == mv && oi < mi)) { mv = ov; mi = oi; }
    }
    if (ln == 0) idx[mtile * 16 + j + ((lane >> 4) << 3)] = mi;
  }
}

// ---------------- gather: h_small[n][c] = emb[idx[n]][c] ----------------
__global__ void vq_gather(const float* __restrict__ emb, const int* __restrict__ idx,
                          float* __restrict__ h_small, int N) {
  int gid = blockIdx.x * blockDim.x + threadIdx.x;
  if (gid >= N * CCH) return;
  int n = gid >> 5, c = gid & 31;
  h_small[gid] = emb[idx[n] * CCH + c];
}

// ---- bilinear upsample h_small [BT,pn,pn,C] -> h_up [BT,32,32,C] (channel-last) ----
__global__ void vq_up(const float* __restrict__ h_small, float* __restrict__ h_up, int pn) {
  int gid = blockIdx.x * blockDim.x + threadIdx.x;
  if (gid >= NUMEL) return;
  int c = gid & 31;
  int p = gid >> 5;                // bt*1024 + y*32 + x
  int x = p & 31, y = (p >> 5) & 31, bt = p >> 10;
  float scale = (float)pn / 32.f;
  float fpn = (float)(pn - 1);
  float sy = fminf(fmaxf((y + 0.5f) * scale - 0.5f, 0.f), fpn);
  float sx = fminf(fmaxf((x + 0.5f) * scale - 0.5f, 0.f), fpn);
  int y0 = (int)sy; int y1 = min(y0 + 1, pn - 1); float wy = sy - (float)y0;
  int x0 = (int)sx; int x1 = min(x0 + 1, pn - 1); float wx = sx - (float)x0;
  const float* base = h_small + (size_t)bt * pn * pn * CCH + c;
  float v00 = base[(y0 * pn + x0) * CCH], v01 = base[(y0 * pn + x1) * CCH];
  float v10 = base[(y1 * pn + x0) * CCH], v11 = base[(y1 * pn + x1) * CCH];
  h_up[gid] = (v00 * (1.f - wy) + v10 * wy) * (1.f - wx) +
              (v01 * (1.f - wy) + v11 * wy) * wx;
}

// ---- WMMA 3x3 conv (implicit im2col) + residual blend + f_hat/f_rest update + loss ----
// grid: 32 frames * 16 tilesets; block 256 = 8 waves; wave -> one (16px x 16ch) tile
__global__ __launch_bounds__(256) void vq_conv(const float* __restrict__ h_up,
                                               const float* __restrict__ phi_w,
                                               const float* __restrict__ phi_b,
                                               const float* __restrict__ f,
                                               float* __restrict__ f_hat,
                                               float* __restrict__ f_rest,
                                               float* __restrict__ loss, int pi) {
  __shared__ float Wl[9 * 32 * 32];   // [tap][k][co]
  __shared__ float Bl[32];
  int tid = threadIdx.x;
  int bt = blockIdx.x >> 4;
  int tileset = blockIdx.x & 15;
  const float* wsrc = phi_w + pi * (CCH * CCH * 9);   // [co][k][tap] flat
  for (int i = tid; i < CCH * CCH * 9; i += 256) {
    int co = i / 288, r = i % 288, k = r / 9, tap = r % 9;
    Wl[tap * 1024 + k * 32 + co] = wsrc[i];
  }
  if (tid < 32) Bl[tid] = phi_b[pi * 32 + tid];
  __syncthreads();

  int wave = tid >> 5, lane = tid & 31;
  int tile = tileset * 8 + wave;        // 0..127 (64 pixel tiles x 2 channel tiles)
  int mtile = tile >> 1, ntile = tile & 1;
  int y = mtile >> 1, xb = (mtile & 1) * 16;
  int ln = lane & 15, koff = (lane >> 4) * 2;
  int co = ntile * 16 + ln;

  v8f acc = {0.f, 0.f, 0.f, 0.f, 0.f, 0.f, 0.f, 0.f};
#pragma unroll
  for (int tap = 0; tap < 9; ++tap) {
    int dy = tap / 3 - 1, dx = tap % 3 - 1;
    int yy = y + dy, xx = xb + ln + dx;
    bool inb = (yy >= 0) && (yy < 32) && (xx >= 0) && (xx < 32);
    const float* src = h_up + (((bt * 32 + yy) * 32 + xx) * CCH) + koff;
#pragma unroll
    for (int s = 0; s < 8; ++s) {
      int k0 = 4 * s + koff;
      v2f aa;
      aa.x = inb ? src[4 * s] : 0.f;
      aa.y = inb ? src[4 * s + 1] : 0.f;
      v2f bb;
      bb.x = Wl[tap * 1024 + k0 * 32 + co];
      bb.y = Wl[tap * 1024 + (k0 + 1) * 32 + co];
      acc = __builtin_amdgcn_wmma_f32_16x16x4_f32(false, aa, false, bb,
                                                  (short)0, acc, false, false);
    }
  }

  // epilogue: y = conv + bias; out = 0.5*h + 0.5*y; update f_hat/f_rest; loss
  float lsum = 0.f;
  int b = bt >> 3, t = bt & 7;
#pragma unroll
  for (int j = 0; j < 8; ++j) {
    int pix = mtile * 16 + j + ((lane >> 4) << 3);     // C-frag row -> pixel
    float h = h_up[((bt * 32 + (pix >> 5)) * 32 + (pix & 31)) * CCH + co];
    float yv = acc[j] + Bl[co];
    float o = 0.5f * h + 0.5f * yv;                    // RESI = 0.5
    int fi = ((b * CCH + co) * 8 + t) * 1024 + pix;
    float fh = f_hat[fi] + o;
    f_hat[fi] = fh;
    f_rest[fi] -= o;
    float d = fh - f[fi];
    lsum += d * d;
  }
#pragma unroll
  for (int off = 16; off > 0; off >>= 1) lsum += __shfl_xor(lsum, off, 32);
  if (lane == 0) atomicAdd(loss, lsum);
}

// ---------------- finalize: d_out = [f_hat, loss] ----------------
__global__ void vq_final(const float* __restrict__ f_hat, const float* __restrict__ loss,
                         float* __restrict__ out) {
  int i = blockIdx.x * blockDim.x + threadIdx.x;
  if (i < NUMEL) out[i] = f_hat[i];
  if (i == 0) out[NUMEL] = loss[0] * (1.25f / (9.f * (float)NUMEL)); // (1+BETA)/SN/numel
}

static inline int cdiv(int a, int b) { return (a + b - 1) / b; }

extern "C" void kernel_launch(void* const* d_in, const int* in_sizes, int n_in,
                              void* d_out, int out_size, void* d_ws, size_t ws_size,
                              hipStream_t stream) {
  const float* f     = (const float*)d_in[0];
  const float* emb   = (const float*)d_in[1];
  const float* phi_w = (const float*)d_in[2];
  const float* phi_b = (const float*)d_in[3];
  float* out = (float*)d_out;

  float* ws      = (float*)d_ws;
  float* f_rest  = ws;
  float* f_hat   = ws + (size_t)NUMEL;
  float* X       = ws + 2 * (size_t)NUMEL;
  float* h_small = ws + 3 * (size_t)NUMEL;
  float* h_up    = ws + 4 * (size_t)NUMEL;
  float* esq     = ws + 5 * (size_t)NUMEL;
  float* lossbuf = esq + NV;
  int*   idxbuf  = (int*)(lossbuf + 1);

  vq_init<<<cdiv(NUMEL, 256), 256, 0, stream>>>(f, f_rest, f_hat, lossbuf);
  vq_esq<<<cdiv(NV, 256), 256, 0, stream>>>(emb, esq);

  static const int PNS[9]  = {1, 2, 4, 6, 9, 13, 18, 24, 32};
  static const int PIDX[9] = {0, 0, 1, 1, 1, 2, 2, 3, 3};

  for (int si = 0; si < 9; ++si) {
    int pn = PNS[si];
    int N  = TFR * pn * pn;
    vq_down<<<cdiv(N * CCH, 256), 256, 0, stream>>>(f_rest, X, pn);
    int Nt = N / 16;
    vq_argmin<<<cdiv(Nt, 8), 256, 0, stream>>>(X, emb, esq, idxbuf, Nt);
    vq_gather<<<cdiv(N * CCH, 256), 256, 0, stream>>>(emb, idxbuf, h_small, N);
    vq_up<<<cdiv(NUMEL, 256), 256, 0, stream>>>(h_small, h_up, pn);
    vq_conv<<<TFR * 16, 256, 0, stream>>>(h_up, phi_w, phi_b, f, f_hat, f_rest,
                                          lossbuf, PIDX[si]);
  }
  vq_final<<<cdiv(NUMEL + 1, 256), 256, 0, stream>>>(f_hat, lossbuf, out);
}